// CodingLoss_26749056320134
// MI455X (gfx1250) — compile-verified
//
#include <hip/hip_runtime.h>
#include <hip/hip_bf16.h>

typedef __attribute__((ext_vector_type(2))) float v2f;
typedef __attribute__((ext_vector_type(8))) float v8f;

#define BB    16384   // batch
#define DD    1000    // feature dim == n_classes
#define NP    1024    // padded N (64 tiles of 16)
#define K2S   2048    // floats per k-pair row of ct2: NP*2
#define ASTR  1002    // LDS stride for A strip (floats)
#define LSTR  1040    // LDS stride for logits (floats)

// ---------------------------------------------------------------------------
// Kernel 1: build ct2[k/2][n][k%2] = code_book[n][k] (zero-padded n>=1000)
//           and csum[n] = sum_k code_book[n][k]  (0 for padded n)
// ---------------------------------------------------------------------------
__global__ void coding_loss_prep(const float* __restrict__ cb,
                                 float* __restrict__ ct2,
                                 float* __restrict__ csum) {
    __shared__ float red[256];
    const int n = blockIdx.x;       // 0..1023
    const int tid = threadIdx.x;
    float s = 0.0f;
    if (n < DD) {
        for (int k = tid; k < DD; k += 256) {
            float v = cb[(size_t)n * DD + k];
            ct2[(size_t)(k >> 1) * K2S + n * 2 + (k & 1)] = v;
            s += v;
        }
    } else {
        for (int k = tid; k < DD; k += 256) {
            ct2[(size_t)(k >> 1) * K2S + n * 2 + (k & 1)] = 0.0f;
        }
    }
    red[tid] = s;
    __syncthreads();
    for (int off = 128; off > 0; off >>= 1) {
        if (tid < off) red[tid] += red[tid + off];
        __syncthreads();
    }
    if (tid == 0) csum[n] = (n < DD) ? red[0] : 0.0f;
}

// ---------------------------------------------------------------------------
// Kernel 2: fused GEMM (fp32 WMMA) + log-softmax + smoothed-CE per row.
// 1 WG = 16 rows of the batch; 8 waves, each wave owns 8 N-tiles of 16.
// ---------------------------------------------------------------------------
__global__ __launch_bounds__(256) void coding_loss_main(
        const float* __restrict__ inputs,
        const int*   __restrict__ labels,
        const float* __restrict__ ct2,
        const float* __restrict__ csum,
        float*       __restrict__ loss_out) {
    __shared__ float smem[16640];   // union: A strip 16x1002 | logits 16x1040

    const int tid  = threadIdx.x;
    const int wave = tid >> 5;
    const int lane = tid & 31;
    const int lh   = lane >> 4;     // half-wave id (0/1)
    const int ll   = lane & 15;
    const int m0   = blockIdx.x * 16;

    // ---- stage A strip (16 x 1000) into LDS, coalesced ----
    for (int r = 0; r < 16; ++r) {
        const float* src = inputs + (size_t)(m0 + r) * DD;
        for (int c = tid; c < DD; c += 256)
            smem[r * ASTR + c] = src[c];
    }
    __syncthreads();

    v8f acc[8];
#pragma unroll
    for (int i = 0; i < 8; ++i) acc[i] = (v8f){0.f,0.f,0.f,0.f,0.f,0.f,0.f,0.f};

    const int ntile0 = wave * 8;    // this wave's first N-tile

    // ---- K loop: D = 1000 = 250 steps of K=4 ----
#pragma unroll 2
    for (int k = 0; k < DD; k += 4) {
        const int kk = k + 2 * lh;  // even
        // A frag: VGPR0 = A[M=ll][kk], VGPR1 = A[M=ll][kk+1]
        v2f a = *(const v2f*)&smem[ll * ASTR + kk];
        const float* brow = ct2 + (size_t)(kk >> 1) * K2S + ll * 2;
#pragma unroll
        for (int i = 0; i < 8; ++i) {
            // B frag: VGPR0 = B[kk][n0+ll], VGPR1 = B[kk+1][n0+ll] (contiguous pair)
            v2f b = *(const v2f*)&brow[(ntile0 + i) * 32];
            acc[i] = __builtin_amdgcn_wmma_f32_16x16x4_f32(
                false, a, false, b, (short)0, acc[i], false, false);
        }
    }
    __syncthreads();   // done reading A strip; reuse LDS for logits

    // ---- write logits L = 2*S - csum[n] into LDS (16 x 1024, stride 1040) ----
#pragma unroll
    for (int i = 0; i < 8; ++i) {
        const int n  = (ntile0 + i) * 16 + ll;
        const float cs = csum[n];
#pragma unroll
        for (int v = 0; v < 8; ++v) {
            const int r = v + 8 * lh;   // C layout: VGPR v, halves -> M = v + 8*lh
            smem[r * LSTR + n] = 2.0f * acc[i][v] - cs;
        }
    }
    __syncthreads();

    // ---- per-row softmax + loss; wave w handles rows 2w, 2w+1 ----
    for (int rr = 0; rr < 2; ++rr) {
        const int r = wave * 2 + rr;
        const float* row = &smem[r * LSTR];
        float mx = -3.402823466e38f;
        for (int n = lane; n < DD; n += 32) mx = fmaxf(mx, row[n]);
#pragma unroll
        for (int off = 16; off > 0; off >>= 1) mx = fmaxf(mx, __shfl_xor(mx, off, 32));
        float se = 0.0f, sl = 0.0f;
        for (int n = lane; n < DD; n += 32) {
            const float x = row[n];
            se += expf(x - mx);
            sl += x;
        }
#pragma unroll
        for (int off = 16; off > 0; off >>= 1) {
            se += __shfl_xor(se, off, 32);
            sl += __shfl_xor(sl, off, 32);
        }
        if (lane == 0) {
            const int lab = labels[m0 + r];
            const float lse = mx + logf(se);
            loss_out[m0 + r] = lse - 0.9f * row[lab] - (0.1f / (float)DD) * sl;
        }
    }
}

// ---------------------------------------------------------------------------
// Kernel 3: deterministic fixed-order reduction of 16384 row losses -> mean
// ---------------------------------------------------------------------------
__global__ void coding_loss_reduce(const float* __restrict__ loss,
                                   float* __restrict__ out) {
    __shared__ float red[256];
    const int tid = threadIdx.x;
    float s = 0.0f;
    for (int i = tid; i < BB; i += 256) s += loss[i];
    red[tid] = s;
    __syncthreads();
    for (int off = 128; off > 0; off >>= 1) {
        if (tid < off) red[tid] += red[tid + off];
        __syncthreads();
    }
    if (tid == 0) out[0] = red[0] * (1.0f / (float)BB);
}

// ---------------------------------------------------------------------------
extern "C" void kernel_launch(void* const* d_in, const int* in_sizes, int n_in,
                              void* d_out, int out_size, void* d_ws, size_t ws_size,
                              hipStream_t stream) {
    const float* inputs = (const float*)d_in[0];   // [16384,1000] f32
    const int*   labels = (const int*)d_in[1];     // [16384] int
    const float* cb     = (const float*)d_in[2];   // [1000,1000] f32
    float* out = (float*)d_out;

    float* ws   = (float*)d_ws;
    float* ct2  = ws;                       // 500*2048      = 1,024,000 floats
    float* csum = ws + 500 * K2S;           // 1024 floats
    float* loss = csum + NP;                // 16384 floats

    coding_loss_prep<<<NP, 256, 0, stream>>>(cb, ct2, csum);
    coding_loss_main<<<BB / 16, 256, 0, stream>>>(inputs, labels, ct2, csum, loss);
    coding_loss_reduce<<<1, 256, 0, stream>>>(loss, out);
}